// KGAN_31310311588010
// MI455X (gfx1250) — compile-verified
//
#include <hip/hip_runtime.h>
#include <math.h>

// Problem constants (match reference)
constexpr int kD  = 32;     // embedding dim
constexpr int kB  = 512;    // batch
constexpr int kR  = 8;      // relation groups per item
constexpr int kM  = 16;     // memories per group
constexpr int kH  = 2;      // hops
constexpr int kNR = 8;      // number of distinct relations (N_REL)

typedef __attribute__((ext_vector_type(2))) float v2f;
typedef __attribute__((ext_vector_type(8))) float v8f;

// 16-byte int vector + address-space-qualified pointers for async-to-LDS DMA
typedef __attribute__((__vector_size__(16))) int i4;
typedef __attribute__((address_space(1))) i4* gi4p;   // global (device)
typedef __attribute__((address_space(3))) i4* li4p;   // LDS (shared)

// ---- WMMA f32 16x16x4 wrapper (CDNA5 V_WMMA_F32_16X16X4_F32) -------------
// 8-arg pattern: (neg_a, A, neg_b, B, c_mod, C, reuse_a, reuse_b)
__device__ __forceinline__ v8f wmma_f32_4(v2f a, v2f b, v8f c) {
  return __builtin_amdgcn_wmma_f32_16x16x4_f32(false, a, false, b, (short)0, c,
                                               false, false);
}

// ---------------------------------------------------------------------------
// k_norms: ||R_k||_F^2 for the 8 relation matrices, and ||T||_F^2.
__global__ void k_norms(const float* __restrict__ relemb,
                        const float* __restrict__ T,
                        float* __restrict__ relnorm, float* __restrict__ tnorm) {
  __shared__ float red[256];
  const float* src = (blockIdx.x < kNR) ? (relemb + (size_t)blockIdx.x * kD * kD) : T;
  float s = 0.f;
  for (int i = threadIdx.x; i < kD * kD; i += 256) { float x = src[i]; s += x * x; }
  red[threadIdx.x] = s;
  __syncthreads();
  if (threadIdx.x == 0) {
    float t = 0.f;
    for (int i = 0; i < 256; ++i) t += red[i];   // fixed order -> deterministic
    if (blockIdx.x < kNR) relnorm[blockIdx.x] = t; else tnorm[0] = t;
  }
}

// ---------------------------------------------------------------------------
// k_initv: v[b,:] = entity_emb[pos_items[b], :]
__global__ void k_initv(const int* __restrict__ pos, const float* __restrict__ ent,
                        float* __restrict__ v) {
  int gid = blockIdx.x * blockDim.x + threadIdx.x;
  if (gid < kB * kD) {
    int b = gid >> 5, d = gid & 31;
    v[gid] = ent[(size_t)pos[b] * kD + d];
  }
}

// ---------------------------------------------------------------------------
// k_gemm_vR: W[b,k,e] = sum_d v[b,d] * R_k[d,e]   (== (R_k^T v_b)_e)
// GEMM V[512x32] @ R_k[32x32] done as WMMA F32 16x16x4 tiles.
__global__ void k_gemm_vR(const float* __restrict__ Vin,
                          const float* __restrict__ relemb,
                          float* __restrict__ W) {
  const int b0   = blockIdx.x * 16;
  const int noff = blockIdx.y * 16;
  const int k    = blockIdx.z;
  const float* Rk = relemb + (size_t)k * kD * kD;
  const int lane = threadIdx.x;
  const int mrow = lane & 15;
  const int hi   = lane >> 4;
  v8f acc = {};
#pragma unroll
  for (int koff = 0; koff < kD; koff += 4) {
    v2f a, bm;
    // A 16x4 layout: VGPR0 = K0 (lanes 0-15) / K2 (lanes 16-31); VGPR1 = K1 / K3
    a.x = Vin[(b0 + mrow) * kD + koff + 2 * hi];
    a.y = Vin[(b0 + mrow) * kD + koff + 2 * hi + 1];
    bm.x = Rk[(koff + 2 * hi) * kD + noff + mrow];
    bm.y = Rk[(koff + 2 * hi + 1) * kD + noff + mrow];
    acc = wmma_f32_4(a, bm, acc);
  }
#pragma unroll
  for (int i = 0; i < 8; ++i) {
    int row = i + 8 * hi;
    W[(b0 + row) * (kNR * kD) + k * kD + noff + mrow] = acc[i];
  }
}

// ---------------------------------------------------------------------------
// k_gemm_vT: v_out[b,e] = sum_d vtmp[b,d] * T[d,e]   ((v+o_hop) @ transform)
__global__ void k_gemm_vT(const float* __restrict__ Vin,
                          const float* __restrict__ T,
                          float* __restrict__ Vout) {
  const int b0   = blockIdx.x * 16;
  const int noff = blockIdx.y * 16;
  const int lane = threadIdx.x;
  const int mrow = lane & 15;
  const int hi   = lane >> 4;
  v8f acc = {};
#pragma unroll
  for (int koff = 0; koff < kD; koff += 4) {
    v2f a, bm;
    a.x  = Vin[(b0 + mrow) * kD + koff + 2 * hi];
    a.y  = Vin[(b0 + mrow) * kD + koff + 2 * hi + 1];
    bm.x = T[(koff + 2 * hi) * kD + noff + mrow];
    bm.y = T[(koff + 2 * hi + 1) * kD + noff + mrow];
    acc = wmma_f32_4(a, bm, acc);
  }
#pragma unroll
  for (int i = 0; i < 8; ++i) {
    int row = i + 8 * hi;
    Vout[(b0 + row) * kD + noff + mrow] = acc[i];
  }
}

// ---------------------------------------------------------------------------
// k_groups: one wave per (b,r) group of 16 memories.
//  - gathers h/t rows into LDS (CDNA5 async-to-LDS DMA when available)
//  - logits via precomputed W[b,k,:]  (v^T R h == (R^T v) . h)
//  - softmax over m, o[b,r,:] = sum_m p_m t_m
//  - kge partial: sum_m sigmoid(h_m^T R_{k_m} t_m)  (2 lanes per memory)
//  - l2 partial: sum h^2 + t^2 + relnorm[k_m]
__global__ void k_groups(int hop,
                         const int* __restrict__ mem_h, const int* __restrict__ mem_r,
                         const int* __restrict__ mem_t,
                         const float* __restrict__ ent, const float* __restrict__ relemb,
                         const float* __restrict__ W, const float* __restrict__ relnorm,
                         float* __restrict__ o_all, float* __restrict__ kge_part,
                         float* __restrict__ l2_part) {
  const int g = blockIdx.x;                 // 0 .. B*R-1
  const int b = g >> 3, r = g & 7;
  const int base = ((hop * kB + b) * kR + r) * kM;
  const int lane = threadIdx.x;

  __shared__ int   hidx[kM], tidx[kM], kidx[kM];
  __shared__ __align__(16) float hs[kM][kD];
  __shared__ __align__(16) float ts[kM][kD];
  __shared__ float lg[kM];
  __shared__ float red[32], red2[32];

  if (lane < kM) {
    hidx[lane] = mem_h[base + lane];
    tidx[lane] = mem_t[base + lane];
    kidx[lane] = mem_r[base + lane];
  }
  __syncthreads();

#if defined(__gfx1250__) && __has_builtin(__builtin_amdgcn_global_load_async_to_lds_b128)
  // CDNA5 async DMA: each lane moves one 16B chunk; 4 rows per issue, 4 issues
  // per tile.  2KB h-tile + 2KB t-tile in 8 instructions, no VGPR round-trip.
  {
    const int arow  = lane >> 3;            // 0..3 (row within issue group)
    const int chunk = lane & 7;             // 16B chunk within 128B row
#pragma unroll
    for (int i = 0; i < 4; ++i) {
      const int rr = i * 4 + arow;
      const float* gh = ent + (size_t)hidx[rr] * kD + chunk * 4;
      const float* gt = ent + (size_t)tidx[rr] * kD + chunk * 4;
      __builtin_amdgcn_global_load_async_to_lds_b128(
          (gi4p)(uintptr_t)gh, (li4p)(uintptr_t)&hs[rr][chunk * 4], 0, 0);
      __builtin_amdgcn_global_load_async_to_lds_b128(
          (gi4p)(uintptr_t)gt, (li4p)(uintptr_t)&ts[rr][chunk * 4], 0, 0);
    }
#if __has_builtin(__builtin_amdgcn_s_wait_asynccnt)
    __builtin_amdgcn_s_wait_asynccnt(0);
#else
    asm volatile("s_wait_asynccnt 0x0" ::: "memory");
#endif
  }
#else
  for (int m = 0; m < kM; ++m) {            // fallback: coalesced row gathers
    hs[m][lane] = ent[(size_t)hidx[m] * kD + lane];
    ts[m][lane] = ent[(size_t)tidx[m] * kD + lane];
  }
#endif
  __syncthreads();

  // attention logits: lane m computes W[b,k_m,:] . h_m  (float4 vectorized)
  if (lane < kM) {
    const float4* wrow = (const float4*)(W + ((size_t)b * kNR + kidx[lane]) * kD);
    const float4* hrow = (const float4*)&hs[lane][0];
    float s = 0.f;
#pragma unroll
    for (int q = 0; q < kD / 4; ++q) {
      float4 wv = wrow[q], hv = hrow[q];
      s += wv.x * hv.x + wv.y * hv.y + wv.z * hv.z + wv.w * hv.w;
    }
    lg[lane] = s;
  }
  __syncthreads();

  float mx = lg[0];
  for (int m = 1; m < kM; ++m) mx = fmaxf(mx, lg[m]);
  float denom = 0.f;
  for (int m = 0; m < kM; ++m) denom += expf(lg[m] - mx);

  float od = 0.f;                           // lane == output dim d
  for (int m = 0; m < kM; ++m) od += expf(lg[m] - mx) * ts[m][lane];
  od /= denom;
  o_all[((size_t)(hop * kB + b) * kR + r) * kD + lane] = od;

  // kge bilinear: lanes (m, m+16) split d-range of h_m^T R t_m.
  // Keep t_m row in registers, stream R_k rows as float4 (global_load_b128).
  const int mrow = lane & 15, hi = lane >> 4;
  const float* Rk = relemb + (size_t)kidx[mrow] * kD * kD;
  const float4* trow = (const float4*)&ts[mrow][0];
  float4 tv[kD / 4];
#pragma unroll
  for (int q = 0; q < kD / 4; ++q) tv[q] = trow[q];
  float part = 0.f;
  for (int d = 16 * hi; d < 16 * hi + 16; ++d) {
    const float4* rrow = (const float4*)(Rk + d * kD);
    float rdot = 0.f;
#pragma unroll
    for (int q = 0; q < kD / 4; ++q) {
      float4 rv = rrow[q];
      rdot += rv.x * tv[q].x + rv.y * tv[q].y + rv.z * tv[q].z + rv.w * tv[q].w;
    }
    part += hs[mrow][d] * rdot;
  }
  red[lane] = part;

  float sq = 0.f;                           // column-sums of h^2 + t^2
  for (int m = 0; m < kM; ++m) { float a = hs[m][lane], c = ts[m][lane]; sq += a * a + c * c; }
  red2[lane] = sq;
  __syncthreads();

  if (lane == 0) {
    float ks = 0.f;
    for (int m = 0; m < kM; ++m) {
      float hrt = red[m] + red[m + 16];
      ks += 1.f / (1.f + expf(-hrt));
    }
    float l2 = 0.f;
    for (int i = 0; i < 32; ++i) l2 += red2[i];
    for (int m = 0; m < kM; ++m) l2 += relnorm[kidx[m]];
    kge_part[hop * (kB * kR) + g] = ks;
    l2_part[hop * (kB * kR) + g]  = l2;
  }
}

// ---------------------------------------------------------------------------
// k_att: one wave per item b.  u = W1 @ w2 folded; att_r = relu(o . u);
// softmax over r; o_hop = sum_r w_r o_r; vtmp = v + o_hop.
__global__ void k_att(int hop, const float* __restrict__ att_w1,
                      const float* __restrict__ att_w2,
                      const float* __restrict__ o_all, const float* __restrict__ v,
                      float* __restrict__ o_hop, float* __restrict__ vtmp) {
  const int b = blockIdx.x;
  const int lane = threadIdx.x;             // == dim d
  __shared__ float red[32];
  __shared__ float att[kR];

  const float* W1 = att_w1 + (size_t)hop * kD * kD;
  const float* w2 = att_w2 + (size_t)hop * kD;
  float u = 0.f;
  for (int e = 0; e < kD; ++e) u += W1[lane * kD + e] * w2[e];

  const float* ob = o_all + ((size_t)(hop * kB + b) * kR) * kD;
  for (int r = 0; r < kR; ++r) {
    red[lane] = ob[r * kD + lane] * u;
    __syncthreads();
    if (lane == 0) {
      float s = 0.f;
      for (int i = 0; i < 32; ++i) s += red[i];
      att[r] = fmaxf(s, 0.f);
    }
    __syncthreads();
  }
  float mx = att[0];
  for (int r = 1; r < kR; ++r) mx = fmaxf(mx, att[r]);
  float denom = 0.f;
  for (int r = 0; r < kR; ++r) denom += expf(att[r] - mx);
  float oh = 0.f;
  for (int r = 0; r < kR; ++r) oh += expf(att[r] - mx) * ob[r * kD + lane];
  oh /= denom;
  o_hop[((size_t)hop * kB + b) * kD + lane] = oh;
  vtmp[b * kD + lane] = v[b * kD + lane] + oh;
}

// ---------------------------------------------------------------------------
// k_final: scores vs ORIGINAL item embeddings, BPR loss, combine kge/l2.
__global__ void k_final(const int* __restrict__ pos, const int* __restrict__ neg,
                        const float* __restrict__ ent,
                        const float* __restrict__ o_hop,
                        const float* __restrict__ kge_part,
                        const float* __restrict__ l2_part,
                        const float* __restrict__ tnorm,
                        float* __restrict__ out) {
  __shared__ float mf[kB], kg[kB], l2s_[kB];
  const int t = threadIdx.x;                // 512 threads, one per item
  const float* ip = ent + (size_t)pos[t] * kD;
  const float* np = ent + (size_t)neg[t] * kD;
  float ps = 0.f, ns = 0.f;
  for (int d = 0; d < kD; ++d) {
    float y = o_hop[t * kD + d] + o_hop[(kB + t) * kD + d];
    ps += ip[d] * y;
    ns += np[d] * y;
  }
  float x  = ps - ns;
  float lg = fminf(x, 0.f) - log1pf(expf(-fabsf(x)));   // stable log-sigmoid
  mf[t] = lg;
  float k = 0.f, l = 0.f;
  for (int i = t; i < kH * kB * kR; i += kB) { k += kge_part[i]; l += l2_part[i]; }
  kg[t] = k; l2s_[t] = l;
  __syncthreads();
  if (t == 0) {
    float mfs = 0.f, kgs = 0.f, l2s = 0.f;
    for (int i = 0; i < kB; ++i) { mfs += mf[i]; kgs += kg[i]; l2s += l2s_[i]; }
    float mf_loss = -mfs / (float)kB;
    float kge     = kgs / (float)(kB * kR * kM);          // sum of per-hop means
    float l2tot   = l2s + (float)kH * 0.5f * tnorm[0];
    out[0] = mf_loss - 0.01f * kge + 1e-5f * l2tot;
  }
}

// ---------------------------------------------------------------------------
extern "C" void kernel_launch(void* const* d_in, const int* in_sizes, int n_in,
                              void* d_out, int out_size, void* d_ws, size_t ws_size,
                              hipStream_t stream) {
  (void)in_sizes; (void)n_in; (void)out_size; (void)ws_size;
  const int*   pos = (const int*)d_in[0];
  const int*   neg = (const int*)d_in[1];
  const int*   mh  = (const int*)d_in[2];
  const int*   mr  = (const int*)d_in[3];
  const int*   mt  = (const int*)d_in[4];
  const float* ent = (const float*)d_in[5];
  const float* rel = (const float*)d_in[6];
  const float* T   = (const float*)d_in[7];
  const float* aw1 = (const float*)d_in[8];
  const float* aw2 = (const float*)d_in[9];
  float* outp = (float*)d_out;

  float* ws = (float*)d_ws;
  float* v        = ws; ws += kB * kD;
  float* vtmp     = ws; ws += kB * kD;
  float* W        = ws; ws += kB * kNR * kD;
  float* o_all    = ws; ws += kH * kB * kR * kD;
  float* o_hop    = ws; ws += kH * kB * kD;
  float* kge_part = ws; ws += kH * kB * kR;
  float* l2_part  = ws; ws += kH * kB * kR;
  float* relnorm  = ws; ws += kNR;
  float* tnorm    = ws; ws += 1;

  hipLaunchKernelGGL(k_norms, dim3(kNR + 1), dim3(256), 0, stream, rel, T, relnorm, tnorm);
  hipLaunchKernelGGL(k_initv, dim3((kB * kD + 255) / 256), dim3(256), 0, stream, pos, ent, v);

  for (int hop = 0; hop < kH; ++hop) {
    hipLaunchKernelGGL(k_gemm_vR, dim3(kB / 16, 2, kNR), dim3(32), 0, stream, v, rel, W);
    hipLaunchKernelGGL(k_groups, dim3(kB * kR), dim3(32), 0, stream, hop, mh, mr, mt,
                       ent, rel, W, relnorm, o_all, kge_part, l2_part);
    hipLaunchKernelGGL(k_att, dim3(kB), dim3(32), 0, stream, hop, aw1, aw2, o_all, v,
                       o_hop, vtmp);
    if (hop + 1 < kH)
      hipLaunchKernelGGL(k_gemm_vT, dim3(kB / 16, 2), dim3(32), 0, stream, vtmp, T, v);
  }

  hipLaunchKernelGGL(k_final, dim3(1), dim3(kB), 0, stream, pos, neg, ent, o_hop,
                     kge_part, l2_part, tnorm, outp);
}